// Entropy_factorized_76940044140584
// MI455X (gfx1250) — compile-verified
//
#include <hip/hip_runtime.h>

#define TILE_ROWS 240
#define NTHREADS  256
#define CCH       32
#define NCONST    43   // per-channel constants: w0[3] B0[3] t0[3] w1[9] B1[3] t1[3] w2[9] B2[3] t2[3] w3[3] B3

typedef unsigned int u32;
typedef unsigned long long u64;

#if defined(__AMDGCN__) && __has_builtin(__builtin_amdgcn_tensor_load_to_lds) && \
    __has_builtin(__builtin_amdgcn_s_wait_tensorcnt)
#define USE_TDM 1
typedef __attribute__((ext_vector_type(4))) u32 tdm_u32x4;
typedef __attribute__((ext_vector_type(8))) int tdm_i32x8;
typedef __attribute__((ext_vector_type(4))) int tdm_i32x4;
#endif

// ---------- fast transcendentals (gfx1250 hardware trans ops) ----------
__device__ __forceinline__ float htanh(float x) {
#if defined(__AMDGCN__) && __has_builtin(__builtin_amdgcn_tanhf)
  return __builtin_amdgcn_tanhf(x);                   // v_tanh_f32
#elif defined(__AMDGCN__)
  float r;                                            // force the hw instruction;
  asm("v_tanh_f32 %0, %1\n\tv_nop"                    // v_nop covers the TRANS
      : "=v"(r) : "v"(x));                            // result-use hazard slot
  return r;
#else
  return tanhf(x);
#endif
}
__device__ __forceinline__ float hexp2(float x) {
#if defined(__AMDGCN__) && __has_builtin(__builtin_amdgcn_exp2f)
  return __builtin_amdgcn_exp2f(x);                   // v_exp_f32
#else
  return exp2f(x);
#endif
}
__device__ __forceinline__ float hlog2(float x) {
#if defined(__AMDGCN__) && __has_builtin(__builtin_amdgcn_logf)
  return __builtin_amdgcn_logf(x);                    // v_log_f32
#else
  return log2f(x);
#endif
}
__device__ __forceinline__ float hrcp(float x) {
#if defined(__AMDGCN__) && __has_builtin(__builtin_amdgcn_rcpf)
  return __builtin_amdgcn_rcpf(x);                    // v_rcp_f32
#else
  return 1.0f / x;
#endif
}
__device__ __forceinline__ float sigmoid_fast(float x) {
  return hrcp(1.0f + hexp2(x * -1.4426950408889634f));  // 1/(1+2^(-x*log2 e))
}
__device__ __forceinline__ float softplusf(float x) {
  return fmaxf(x, 0.0f) + log1pf(expf(-fabsf(x)));    // precise; precompute-only path
}

#ifdef USE_TDM
// 1-D contiguous TDM copy: global -> LDS, `elems` fp32 elements (elems <= 65535).
__device__ __forceinline__ void tdm_load_1d(u32 lds_byte_off, const float* gsrc, u32 elems) {
  u64 ga = (u64)(size_t)gsrc;
  tdm_u32x4 g0;
  g0[0] = 1u;                                            // count=1, user descriptor
  g0[1] = lds_byte_off;                                  // LDS byte address
  g0[2] = (u32)ga;                                       // global_addr[31:0]
  g0[3] = (u32)((ga >> 32) & 0x1FFFFFFull) | (2u << 30); // global_addr[56:32] | type=2
  tdm_i32x8 g1;
  g1[0] = (int)(2u << 16);                               // data_size = 4 bytes
  g1[1] = (int)((elems & 0xFFFFu) << 16);                // tensor_dim0[15:0]
  g1[2] = (int)(((elems >> 16) & 0xFFFFu) | (1u << 16)); // tensor_dim0[31:16], tensor_dim1=1
  g1[3] = (int)((elems & 0xFFFFu) << 16);                // tile_dim0 = elems
  g1[4] = 1;                                             // tile_dim1=1, tile_dim2=0
  g1[5] = (int)elems;                                    // tensor_dim0_stride[31:0]
  g1[6] = 0;
  g1[7] = 0;
  tdm_i32x4 gz4 = {0, 0, 0, 0};                          // groups 2/3 unused (<=2D)
  tdm_i32x8 gz8 = {0, 0, 0, 0, 0, 0, 0, 0};
  __builtin_amdgcn_tensor_load_to_lds(g0, g1, gz4, gz4, gz8, 0);
}
#endif

// ---- tiny prep kernel: per-channel constants -> d_ws, layout [NCONST][32] ----
__global__ void entropy_prep_kernel(const float* __restrict__ m0, const float* __restrict__ m1,
                                    const float* __restrict__ m2, const float* __restrict__ m3,
                                    const float* __restrict__ b0, const float* __restrict__ b1,
                                    const float* __restrict__ b2, const float* __restrict__ b3,
                                    const float* __restrict__ f0, const float* __restrict__ f1,
                                    const float* __restrict__ f2, float* __restrict__ cw) {
  int c = (int)threadIdx.x;
  if (c >= CCH) return;
#pragma unroll
  for (int j = 0; j < 3; ++j) {
    cw[(0 + j) * CCH + c]  = softplusf(m0[c * 3 + j]);
    cw[(3 + j) * CCH + c]  = b0[c * 3 + j];
    cw[(6 + j) * CCH + c]  = tanhf(f0[c * 3 + j]);
    cw[(18 + j) * CCH + c] = b1[c * 3 + j];
    cw[(21 + j) * CCH + c] = tanhf(f1[c * 3 + j]);
    cw[(33 + j) * CCH + c] = b2[c * 3 + j];
    cw[(36 + j) * CCH + c] = tanhf(f2[c * 3 + j]);
    cw[(39 + j) * CCH + c] = softplusf(m3[c * 3 + j]);
  }
#pragma unroll
  for (int j = 0; j < 9; ++j) {
    cw[(9 + j) * CCH + c]  = softplusf(m1[c * 9 + j]);
    cw[(24 + j) * CCH + c] = softplusf(m2[c * 9 + j]);
  }
  cw[42 * CCH + c] = b3[c];
}

__global__ __launch_bounds__(NTHREADS)
void entropy_bits_kernel(const float* __restrict__ x,
                         const float* __restrict__ m0, const float* __restrict__ m1,
                         const float* __restrict__ m2, const float* __restrict__ m3,
                         const float* __restrict__ b0, const float* __restrict__ b1,
                         const float* __restrict__ b2, const float* __restrict__ b3,
                         const float* __restrict__ f0, const float* __restrict__ f1,
                         const float* __restrict__ f2,
                         const float* __restrict__ cw, int use_ws,
                         float* __restrict__ out, int N, int ntiles) {
  const int c = (int)(threadIdx.x & (CCH - 1));   // lane -> channel

  float w0[3], B0[3], t0[3];
  float w1[9], B1[3], t1[3];
  float w2[9], B2[3], t2[3];
  float w3[3], B3;

  if (use_ws) {   // fast path: 43 coalesced loads of precomputed constants
#pragma unroll
    for (int j = 0; j < 3; ++j) {
      w0[j] = cw[(0 + j) * CCH + c];  B0[j] = cw[(3 + j) * CCH + c];  t0[j] = cw[(6 + j) * CCH + c];
      B1[j] = cw[(18 + j) * CCH + c]; t1[j] = cw[(21 + j) * CCH + c];
      B2[j] = cw[(33 + j) * CCH + c]; t2[j] = cw[(36 + j) * CCH + c];
      w3[j] = cw[(39 + j) * CCH + c];
    }
#pragma unroll
    for (int j = 0; j < 9; ++j) { w1[j] = cw[(9 + j) * CCH + c]; w2[j] = cw[(24 + j) * CCH + c]; }
    B3 = cw[42 * CCH + c];
  } else {        // fallback: compute in-kernel
#pragma unroll
    for (int j = 0; j < 3; ++j) {
      w0[j] = softplusf(m0[c * 3 + j]);
      B0[j] = b0[c * 3 + j];  t0[j] = tanhf(f0[c * 3 + j]);
      B1[j] = b1[c * 3 + j];  t1[j] = tanhf(f1[c * 3 + j]);
      B2[j] = b2[c * 3 + j];  t2[j] = tanhf(f2[c * 3 + j]);
      w3[j] = softplusf(m3[c * 3 + j]);
    }
#pragma unroll
    for (int j = 0; j < 9; ++j) { w1[j] = softplusf(m1[c * 9 + j]); w2[j] = softplusf(m2[c * 9 + j]); }
    B3 = b3[c];
  }

  auto logits = [&](float v) -> float {
    float h[3], g[3];
#pragma unroll
    for (int j = 0; j < 3; ++j) {                  // layer0 (1->3) + factor
      float t = fmaf(w0[j], v, B0[j]);
      h[j] = fmaf(t0[j], htanh(t), t);
    }
#pragma unroll
    for (int j = 0; j < 3; ++j) {                  // layer1 (3->3) + factor
      float t = fmaf(w1[j * 3 + 0], h[0], fmaf(w1[j * 3 + 1], h[1], fmaf(w1[j * 3 + 2], h[2], B1[j])));
      g[j] = fmaf(t1[j], htanh(t), t);
    }
#pragma unroll
    for (int j = 0; j < 3; ++j) {                  // layer2 (3->3) + factor
      float t = fmaf(w2[j * 3 + 0], g[0], fmaf(w2[j * 3 + 1], g[1], fmaf(w2[j * 3 + 2], g[2], B2[j])));
      h[j] = fmaf(t2[j], htanh(t), t);
    }
    return fmaf(w3[0], h[0], fmaf(w3[1], h[1], fmaf(w3[2], h[2], B3)));  // layer3 (3->1)
  };

  auto eval_bits = [&](float v) -> float {
    float lo = logits(v - 0.5f);
    float up = logits(v + 0.5f);
    float s  = lo + up;
    float sg = (s > 0.0f) ? -1.0f : ((s < 0.0f) ? 1.0f : 0.0f);   // -sign(lo+up)
    float like = fabsf(sigmoid_fast(sg * up) - sigmoid_fast(sg * lo));
    like = fmaxf(like, 1e-6f);
    return -hlog2(like);
  };

#ifdef USE_TDM
  // ---- TDM double-buffered LDS staging (wave 0 drives the DMA) ----
  __shared__ float lbuf[2][TILE_ROWS * CCH];                // 2 x 30 KB
  const u32 buf_bytes = (u32)(TILE_ROWS * CCH * 4);
  const u32 lds0 = (u32)(u64)(size_t)(&lbuf[0][0]);         // low 32 bits = LDS offset
  const int wave0 = (__builtin_amdgcn_readfirstlane((int)(threadIdx.x >> 5)) == 0);

  int t = (int)blockIdx.x;
  int k = 0;
  if (t < ntiles && wave0) {
    int base = t * TILE_ROWS;
    int rows = N - base; if (rows > TILE_ROWS) rows = TILE_ROWS;
    tdm_load_1d(lds0, x + (u64)base * CCH, (u32)rows * CCH);
  }
  for (; t < ntiles; t += (int)gridDim.x, k ^= 1) {
    int tn = t + (int)gridDim.x;
    if (wave0) {
      if (tn < ntiles) {                                    // prefetch next tile
        int nbase = tn * TILE_ROWS;
        int nrows = N - nbase; if (nrows > TILE_ROWS) nrows = TILE_ROWS;
        tdm_load_1d(lds0 + (u32)(k ^ 1) * buf_bytes, x + (u64)nbase * CCH, (u32)nrows * CCH);
        __builtin_amdgcn_s_wait_tensorcnt(1);               // current tile done
      } else {
        __builtin_amdgcn_s_wait_tensorcnt(0);
      }
    }
    __syncthreads();                                        // tile t visible to all waves
    int base = t * TILE_ROWS;
    int rows = N - base; if (rows > TILE_ROWS) rows = TILE_ROWS;
    const float* buf = &lbuf[k][0];
    for (int r = (int)(threadIdx.x >> 5); r < rows; r += NTHREADS / CCH) {
      float v = buf[r * CCH + c];
      out[(u64)(base + r) * CCH + c] = eval_bits(v);
    }
    __syncthreads();                                        // buffer free before overwrite
  }
#else
  // ---- fallback: direct coalesced global path ----
  u32 total  = (u32)N * CCH;
  u32 stride = gridDim.x * NTHREADS;
  for (u32 i = blockIdx.x * NTHREADS + threadIdx.x; i < total; i += stride) {
    out[i] = eval_bits(x[i]);   // i % 32 == threadIdx.x % 32 (stride multiple of 32)
  }
#endif
}

extern "C" void kernel_launch(void* const* d_in, const int* in_sizes, int n_in,
                              void* d_out, int out_size, void* d_ws, size_t ws_size,
                              hipStream_t stream) {
  const float* x  = (const float*)d_in[0];
  const float* m0 = (const float*)d_in[1];
  const float* m1 = (const float*)d_in[2];
  const float* m2 = (const float*)d_in[3];
  const float* m3 = (const float*)d_in[4];
  const float* b0 = (const float*)d_in[5];
  const float* b1 = (const float*)d_in[6];
  const float* b2 = (const float*)d_in[7];
  const float* b3 = (const float*)d_in[8];
  const float* f0 = (const float*)d_in[9];
  const float* f1 = (const float*)d_in[10];
  const float* f2 = (const float*)d_in[11];

  int N      = in_sizes[0] / CCH;
  int ntiles = (N + TILE_ROWS - 1) / TILE_ROWS;
  int blocks = ntiles < 1024 ? ntiles : 1024;   // persistent blocks -> TDM pipelining
  if (blocks < 1) blocks = 1;

  int use_ws = (ws_size >= (size_t)(NCONST * CCH * sizeof(float))) ? 1 : 0;
  if (use_ws) {
    hipLaunchKernelGGL(entropy_prep_kernel, dim3(1), dim3(CCH), 0, stream,
                       m0, m1, m2, m3, b0, b1, b2, b3, f0, f1, f2, (float*)d_ws);
  }
  hipLaunchKernelGGL(entropy_bits_kernel, dim3(blocks), dim3(NTHREADS), 0, stream,
                     x, m0, m1, m2, m3, b0, b1, b2, b3, f0, f1, f2,
                     (const float*)d_ws, use_ws, (float*)d_out, N, ntiles);
}